// OverlapMeanStdTripletLoss_11991548690926
// MI455X (gfx1250) — compile-verified
//
#include <hip/hip_runtime.h>
#include <math.h>

typedef float v2f __attribute__((ext_vector_type(2)));
typedef float v8f __attribute__((ext_vector_type(8)));

#define NPTS  65536
#define DIMS  256
#define NCLS  16
#define KSEL  2048
#define NMAX  8192
#define ALPHAv 1.0f
#define EPSv   1e-6f

// ---------------- init: zero accumulators ----------------
__global__ __launch_bounds__(256) void k_init(float* csum, unsigned* gcount) {
    int t = blockIdx.x * 256 + threadIdx.x;
    if (t < NCLS) gcount[t] = 0u;
    for (int k = t; k < NCLS * DIMS; k += gridDim.x * 256) csum[k] = 0.f;
}

// ---------------- pass 1: center sums, counts, slots, |x|^2 ----------------
__global__ __launch_bounds__(256) void k_stats(const float* __restrict__ x,
                                               const int* __restrict__ y,
                                               float* __restrict__ csum,
                                               unsigned* __restrict__ gcount,
                                               float* __restrict__ xnorm2,
                                               int* __restrict__ slotArr) {
    __shared__ float lsum[NCLS * DIMS];
    int tid = threadIdx.x;
    for (int t = tid; t < NCLS * DIMS; t += 256) lsum[t] = 0.f;
    __syncthreads();
    int wave = tid >> 5, lane = tid & 31;
    for (int r = wave; r < 256; r += 8) {  // one wave per row, coalesced
        int n = blockIdx.x * 256 + r;
        int L = y[3 * n + 2] & (NCLS - 1);
        const float* row = x + (size_t)n * DIMS;
        float vals[8];
        float sq = 0.f;
#pragma unroll
        for (int j = 0; j < 8; ++j) { float v = row[lane + 32 * j]; vals[j] = v; sq += v * v; }
#pragma unroll
        for (int off = 16; off > 0; off >>= 1) sq += __shfl_down(sq, off, 32);
        if (lane == 0) {
            xnorm2[n] = sq;
            unsigned s = atomicAdd(&gcount[L], 1u);
            slotArr[n] = (int)s;
        }
#pragma unroll
        for (int j = 0; j < 8; ++j) atomicAdd(&lsum[L * DIMS + lane + 32 * j], vals[j]);
    }
    __syncthreads();
    for (int t = tid; t < NCLS * DIMS; t += 256) atomicAdd(&csum[t], lsum[t]);
}

// ---------------- finalize centers + |c|^2 ----------------
__global__ __launch_bounds__(256) void k_centers(const float* __restrict__ csum,
                                                 const unsigned* __restrict__ gcount,
                                                 float* __restrict__ centers,
                                                 float* __restrict__ cnorm2) {
    __shared__ float red[256];
    int i = blockIdx.x, k = threadIdx.x;
    float cnt = (float)gcount[i];
    float v = csum[i * DIMS + k] / cnt + EPSv;
    centers[i * DIMS + k] = v;
    red[k] = v * v;
    __syncthreads();
    for (int s = 128; s > 0; s >>= 1) { if (k < s) red[k] += red[k + s]; __syncthreads(); }
    if (k == 0) cnorm2[i] = red[0];
}

// ---------------- pass 2: D = sqrt(|c|^2 - 2 c.x + |x|^2) via f32 WMMA,
//                  scatter into per-(class,row) gather buffers ----------------
__global__ __launch_bounds__(128) void k_dist(const float* __restrict__ x,
                                              const int* __restrict__ y,
                                              const float* __restrict__ centers,
                                              const float* __restrict__ cnorm2,
                                              const float* __restrict__ xnorm2,
                                              const int* __restrict__ slotArr,
                                              float* __restrict__ G) {
    int wave = threadIdx.x >> 5, lane = threadIdx.x & 31;
    int tile = blockIdx.x * 4 + wave;         // 4096 tiles of 16 x-rows
    int n0 = tile * 16;
    int mrow  = lane & 15;                    // M for A, N for B (same striping)
    int khalf = (lane >> 4) << 1;             // lanes 16-31 hold K+2,K+3
    const float* arow = centers + (size_t)mrow * DIMS + khalf;
    const float* brow = x + (size_t)(n0 + mrow) * DIMS + khalf;

    v8f acc = {};
#pragma unroll 4
    for (int kb = 0; kb < DIMS; kb += 4) {
        v2f a = *(const v2f*)(arow + kb);     // A[mrow, kb+khalf .. +1]
        v2f b = *(const v2f*)(brow + kb);     // B[kb+khalf.., n0+mrow]
        acc = __builtin_amdgcn_wmma_f32_16x16x4_f32(
            false, a, false, b, (short)0, acc, false, false);
    }

    int n = n0 + mrow;
    float xn  = xnorm2[n];
    int  L    = y[3 * n + 2] & (NCLS - 1);
    int  slot = slotArr[n];
    int  ibase = (lane >> 4) << 3;            // C layout: VGPR r -> M = r + 8*(lane/16)
    if (slot < NMAX) {
#pragma unroll
        for (int r = 0; r < 8; ++r) {
            int i = ibase + r;
            float d2 = cnorm2[i] - 2.f * acc[r] + xn;
            float D = sqrtf(fmaxf(d2, 0.f));
            G[((size_t)(L * NCLS + i)) * NMAX + slot] = D;
        }
    }
}

// ---------------- per-(i,c) bitonic sort in LDS + top/bottom-K stats ----------------
__global__ __launch_bounds__(1024) void k_select(const float* __restrict__ G,
                                                 const unsigned* __restrict__ gcount,
                                                 float* __restrict__ posStat,
                                                 float* __restrict__ negStat) {
    __shared__ float arr[NMAX];
    __shared__ float redA[1024];
    __shared__ float redB[1024];
    int tid  = threadIdx.x;
    int pair = blockIdx.x;            // pair = c*16 + i (matches gather layout)
    int c = pair >> 4;
    int i = pair & 15;
    int cnt = min((int)gcount[c], NMAX);
    const float* src = G + (size_t)pair * NMAX;
    for (int t = tid; t < NMAX; t += 1024)
        arr[t] = (t < cnt) ? src[t] : __builtin_huge_valf();
    __syncthreads();

    // bitonic sort ascending (pads = +inf at the end)
    for (unsigned k = 2; k <= NMAX; k <<= 1) {
        for (unsigned j = k >> 1; j > 0; j >>= 1) {
            for (unsigned idx = tid; idx < NMAX; idx += 1024) {
                unsigned ixj = idx ^ j;
                if (ixj > idx) {
                    float a = arr[idx], b = arr[ixj];
                    bool up = ((idx & k) == 0);
                    if ((a > b) == up) { arr[idx] = b; arr[ixj] = a; }
                }
            }
            __syncthreads();
        }
    }

    // neg stats: smallest KSEL = arr[0 .. KSEL)
    {
        float a0 = arr[tid], a1 = arr[tid + 1024];
        redA[tid] = a0 + a1;
        redB[tid] = a0 * a0 + a1 * a1;
        __syncthreads();
        for (int s = 512; s > 0; s >>= 1) {
            if (tid < s) { redA[tid] += redA[tid + s]; redB[tid] += redB[tid + s]; }
            __syncthreads();
        }
        if (tid == 0) {
            float mean = redA[0] / (float)KSEL;
            float var  = (redB[0] - (float)KSEL * mean * mean) / (float)(KSEL - 1);
            negStat[i * NCLS + c] = mean - 1.96f * sqrtf(fmaxf(var, 0.f));
        }
        __syncthreads();
    }

    // pos stats (diag only): largest KSEL members = arr[cnt-KSEL .. cnt)
    if (i == c) {
        int base = max(cnt - KSEL, 0);
        float a0 = arr[base + tid], a1 = arr[base + tid + 1024];
        redA[tid] = a0 + a1;
        redB[tid] = a0 * a0 + a1 * a1;
        __syncthreads();
        for (int s = 512; s > 0; s >>= 1) {
            if (tid < s) { redA[tid] += redA[tid + s]; redB[tid] += redB[tid + s]; }
            __syncthreads();
        }
        if (tid == 0) {
            float mean = redA[0] / (float)KSEL;
            float var  = (redB[0] - (float)KSEL * mean * mean) / (float)(KSEL - 1);
            posStat[i] = mean + 1.96f * sqrtf(fmaxf(var, 0.f));
        }
    }
}

// ---------------- final 256-term hinge reduction ----------------
__global__ __launch_bounds__(256) void k_loss(const float* __restrict__ posStat,
                                              const float* __restrict__ negStat,
                                              float* __restrict__ out) {
    __shared__ float red[256];
    int t = threadIdx.x;
    int i = t >> 4, c = t & 15;
    float term = 0.f;
    if (i != c) term = fmaxf(ALPHAv + posStat[i] - negStat[t], 0.f);
    red[t] = term;
    __syncthreads();
    for (int s = 128; s > 0; s >>= 1) { if (t < s) red[t] += red[t + s]; __syncthreads(); }
    if (t == 0) out[0] = red[0];
}

extern "C" void kernel_launch(void* const* d_in, const int* in_sizes, int n_in,
                              void* d_out, int out_size, void* d_ws, size_t ws_size,
                              hipStream_t stream) {
    (void)in_sizes; (void)n_in; (void)out_size; (void)ws_size;
    const float* x = (const float*)d_in[0];
    const int*   y = (const int*)d_in[1];

    char* ws = (char*)d_ws;
    size_t off = 0;
    auto take = [&](size_t bytes) -> char* {
        char* p = ws + off;
        off = (off + bytes + 255) & ~(size_t)255;
        return p;
    };
    unsigned* gcount  = (unsigned*)take(NCLS * sizeof(unsigned));
    float*    csum    = (float*)take((size_t)NCLS * DIMS * sizeof(float));
    float*    centers = (float*)take((size_t)NCLS * DIMS * sizeof(float));
    float*    cnorm2  = (float*)take(NCLS * sizeof(float));
    float*    xnorm2  = (float*)take((size_t)NPTS * sizeof(float));
    int*      slotArr = (int*)take((size_t)NPTS * sizeof(int));
    float*    posStat = (float*)take(NCLS * sizeof(float));
    float*    negStat = (float*)take(NCLS * NCLS * sizeof(float));
    float*    G       = (float*)take((size_t)NCLS * NCLS * NMAX * sizeof(float));

    k_init   <<<16, 256, 0, stream>>>(csum, gcount);
    k_stats  <<<NPTS / 256, 256, 0, stream>>>(x, y, csum, gcount, xnorm2, slotArr);
    k_centers<<<NCLS, 256, 0, stream>>>(csum, gcount, centers, cnorm2);
    k_dist   <<<NPTS / 16 / 4, 128, 0, stream>>>(x, y, centers, cnorm2, xnorm2, slotArr, G);
    k_select <<<NCLS * NCLS, 1024, 0, stream>>>(G, gcount, posStat, negStat);
    k_loss   <<<1, 256, 0, stream>>>(posStat, negStat, (float*)d_out);
}